// ConvectionM2_18880676233727
// MI455X (gfx1250) — compile-verified
//
#include <hip/hip_runtime.h>

#define BB    4
#define CC    16
#define OR    8
#define HH    256
#define WW    256
#define ROWS  16            // output rows per workgroup
#define NROWS (ROWS + 1)    // staged input rows per plane
#define PITCH 260           // floats per LDS row (16B-aligned pitch, shifts banks)

#if defined(__gfx1250__) && __has_builtin(__builtin_amdgcn_global_load_async_to_lds_b128)
#define USE_ASYNC_LDS 1
#else
#define USE_ASYNC_LDS 0
#endif

typedef int v4i __attribute__((vector_size(16)));
typedef __attribute__((address_space(1))) v4i* gptr_v4i;
typedef __attribute__((address_space(3))) v4i* lptr_v4i;

__global__ __launch_bounds__(256)
void m2_convection_kernel(const float* __restrict__ x,
                          const float* __restrict__ g0,
                          float* __restrict__ out)
{
    __shared__ float lds[2 * NROWS * PITCH];

    const int tid = threadIdx.x;
    const int i0  = blockIdx.x * ROWS;
    const int ck  = blockIdx.y;          // c*OR + k
    const int c   = ck >> 3;
    const int k   = ck & 7;
    const int b   = blockIdx.z;

    // ---- per-(c,k) scalar setup (uniform across the block) ----
    const float x0  = g0[c * 3 + 0];
    const float y0  = g0[c * 3 + 1];
    const float th0 = g0[c * 3 + 2];

    const float TWO_PI = 6.28318530717958647692f;
    const float alpha  = (float)k * (TWO_PI / (float)OR) - th0;
    const float ca = cosf(alpha);
    const float sa = sinf(alpha);
    const float dxs = ca * x0 - sa * y0;
    const float dys = sa * x0 + ca * y0;

    const float t   = (float)k - th0 * ((float)OR / TWO_PI);
    const float tf  = floorf(t);
    const float ft  = t - tf;
    const int   t0i = (int)tf;
    const int   ki0 = t0i & 7;           // mod 8, correct for negatives (pow2)
    const int   ki1 = (ki0 + 1) & 7;
    const float wt0 = 1.0f - ft, wt1 = ft;

    const float mdx = -dxs;
    const float fxf = floorf(mdx);
    const float fx  = mdx - fxf;
    const int   ixb = (int)fxf;
    const float wx0 = 1.0f - fx, wx1 = fx;

    const float mdy = -dys;
    const float fyf = floorf(mdy);
    const float fy  = mdy - fyf;
    const int   iyb = (int)fyf;
    const float wy0 = 1.0f - fy, wy1 = fy;

    const size_t planeSz = (size_t)HH * WW;
    const float* p0 = x + (((size_t)b * CC + c) * OR + ki0) * planeSz;
    const float* p1 = x + (((size_t)b * CC + c) * OR + ki1) * planeSz;

    // ---- stage 2 planes x 17 rows into LDS (edge rows clamped; weights do the zeroing) ----
    const int rbase = i0 + iyb;
    const int TOT4  = 2 * NROWS * (WW / 4);          // float4 elements to stage
    for (int idx = tid; idx < TOT4; idx += 256) {
        const int plane = idx / (NROWS * (WW / 4));
        const int rem   = idx - plane * (NROWS * (WW / 4));
        const int lrow  = rem / (WW / 4);
        const int c4    = rem - lrow * (WW / 4);
        int r = rbase + lrow;
        r = r < 0 ? 0 : (r > HH - 1 ? HH - 1 : r);
        const float* gsrc = (plane ? p1 : p0) + (size_t)r * WW + c4 * 4;
        float*       ldst = &lds[(plane * NROWS + lrow) * PITCH + c4 * 4];
#if USE_ASYNC_LDS
        __builtin_amdgcn_global_load_async_to_lds_b128(
            (gptr_v4i)gsrc, (lptr_v4i)ldst, 0, 0);
#else
        *(float4*)ldst = *(const float4*)gsrc;
#endif
    }
#if USE_ASYNC_LDS
#if __has_builtin(__builtin_amdgcn_s_wait_asynccnt)
    __builtin_amdgcn_s_wait_asynccnt(0);
#else
    asm volatile("s_wait_asynccnt 0" ::: "memory");
#endif
#endif
    __syncthreads();

    // ---- compute: one column per thread, 16 rows, rolling two-row reuse ----
    const int j  = tid;
    const int xA = j + ixb;
    const int xB = xA + 1;
    const float wxA = wx0 * ((xA >= 0 && xA < WW) ? 1.0f : 0.0f);
    const float wxB = wx1 * ((xB >= 0 && xB < WW) ? 1.0f : 0.0f);
    const int   iA  = xA < 0 ? 0 : (xA > WW - 1 ? WW - 1 : xA);
    const int   iB  = xB < 0 ? 0 : (xB > WW - 1 ? WW - 1 : xB);

    const float* l0 = &lds[0];
    const float* l1 = &lds[NROWS * PITCH];

    float tA = wt0 * l0[iA] + wt1 * l1[iA];
    float tB = wt0 * l0[iB] + wt1 * l1[iB];

    float* obase = out + ((((size_t)b * CC + c) * OR + k) * HH + i0) * WW + j;

    #pragma unroll
    for (int l = 0; l < ROWS; ++l) {
        const int yA = rbase + l;
        const int yB = yA + 1;
        const float wA = wy0 * ((yA >= 0 && yA < HH) ? 1.0f : 0.0f);
        const float wB = wy1 * ((yB >= 0 && yB < HH) ? 1.0f : 0.0f);
        const int   ro = (l + 1) * PITCH;
        const float bA = wt0 * l0[ro + iA] + wt1 * l1[ro + iA];
        const float bB = wt0 * l0[ro + iB] + wt1 * l1[ro + iB];
        const float val = wA * (wxA * tA + wxB * tB) + wB * (wxA * bA + wxB * bB);
        __builtin_nontemporal_store(val, obase + (size_t)l * WW);   // keep 192MB L2 for the input
        tA = bA; tB = bB;
    }
}

extern "C" void kernel_launch(void* const* d_in, const int* in_sizes, int n_in,
                              void* d_out, int out_size, void* d_ws, size_t ws_size,
                              hipStream_t stream) {
    (void)in_sizes; (void)n_in; (void)d_ws; (void)ws_size; (void)out_size;
    const float* x  = (const float*)d_in[0];
    const float* g0 = (const float*)d_in[1];
    float* out = (float*)d_out;
    dim3 grid(HH / ROWS, CC * OR, BB);   // 16 x 128 x 4 = 8192 workgroups
    m2_convection_kernel<<<grid, 256, 0, stream>>>(x, g0, out);
}